// Min_interval_66640712564831
// MI455X (gfx1250) — compile-verified
//
#include <hip/hip_runtime.h>
#include <stdint.h>

#define NCOLS 16
#define NL2   120          // C(16,2)
#define NL3   560          // C(16,3)
#define PRE   136          // 16 + 120 prefix kept in LDS
#define OUTC  696          // 16 + 120 + 560
#define NG4   174          // OUTC / 4 column groups
#define RPB   8            // rows per block
#define TPB   256

typedef float v4f __attribute__((ext_vector_type(4)));

// ---- compile-time gather tables (exactly mirrors itertools.combinations order)
struct Tables {
    uint32_t lvl2[NL2];    // a | (b<<5)   : level-2 from raw cols (l=x[b], r=x[a])
    uint32_t lvl3[NL3];    // il | (ir<<8) : indices into 136-entry prefix
};

static constexpr Tables make_tables() {
    Tables T{};
    int pidx[16][16] = {};
    int k = 0;
    for (int a = 0; a < NCOLS; ++a)
        for (int b = a + 1; b < NCOLS; ++b) {
            T.lvl2[k] = (uint32_t)a | ((uint32_t)b << 5);
            pidx[a][b] = 16 + k;            // global index of subset (a,b)
            ++k;
        }
    k = 0;
    for (int a = 0; a < NCOLS; ++a)
        for (int b = a + 1; b < NCOLS; ++b)
            for (int c = b + 1; c < NCOLS; ++c) {
                uint32_t il = (uint32_t)pidx[b][c];   // left  = res[c[1:]]
                uint32_t ir = (uint32_t)pidx[a][b];   // right = res[c[:-1]]
                T.lvl3[k++] = il | (ir << 8);
            }
    return T;
}
__constant__ const Tables g_tab = make_tables();

// ---- admissible min under K_{alpha,beta}; contraction off so ka/kb round
// exactly like the reference's unfused mul-then-add (tie-breaks are exact
// float compares).
__device__ __forceinline__ void admin(float ll, float lu, float rl, float ru,
                                      float& ol, float& ou) {
#pragma clang fp contract(off)
    float ka_l = ll + 0.4f * (lu - ll);
    float ka_r = rl + 0.4f * (ru - rl);
    float kb_l = ll + 0.6f * (lu - ll);
    float kb_r = rl + 0.6f * (ru - rl);
    bool take_r = (ka_r < ka_l) || ((ka_r == ka_l) && (kb_r <= kb_l));
    ol = take_r ? rl : ll;
    ou = take_r ? ru : lu;
}

__global__ __launch_bounds__(TPB) void
min_interval_kernel(const float* __restrict__ xl, const float* __restrict__ xu,
                    float* __restrict__ outl, float* __restrict__ outu, int batch) {
    // Per-row 136-entry prefix (16 raw + 120 level-2), 16B-aligned rows for b128.
    __shared__ __attribute__((aligned(16))) float s_l[RPB][PRE];
    __shared__ __attribute__((aligned(16))) float s_u[RPB][PRE];

    const int t    = threadIdx.x;
    const int row0 = blockIdx.x * RPB;

    // ---- Stage raw columns via gfx1250 async global->LDS (ASYNCcnt path).
    // 8 rows x 16 cols = 128 dwords per buffer; waves 0..3 issue, one dword/lane.
    if (t < RPB * NCOLS) {
        const int r = t >> 4, cc = t & 15;
        const int row = row0 + r;
        if (row < batch) {
            uint32_t goff = (uint32_t)(row * NCOLS + cc) * 4u;
            uint32_t adl  = (uint32_t)(uintptr_t)&s_l[r][cc];  // low 32b == LDS offset
            uint32_t adu  = (uint32_t)(uintptr_t)&s_u[r][cc];
            asm volatile("global_load_async_to_lds_b32 %0, %1, %2"
                         :: "v"(adl), "v"(goff), "s"(xl) : "memory");
            asm volatile("global_load_async_to_lds_b32 %0, %1, %2"
                         :: "v"(adu), "v"(goff), "s"(xu) : "memory");
        }
    }
    asm volatile("s_wait_asynccnt 0" ::: "memory");
    __syncthreads();

    // ---- Phase 1: level-2 prefix. 8 rows x 120 subsets = 960 tasks.
    for (int w = t; w < RPB * NL2; w += TPB) {
        int r = w / NL2, j = w - r * NL2;
        uint32_t e = g_tab.lvl2[j];
        int a = e & 31, b = (e >> 5) & 31;
        float ol, ou;
        admin(s_l[r][b], s_u[r][b], s_l[r][a], s_u[r][a], ol, ou);  // l=x[b], r=x[a]
        s_l[r][16 + j] = ol;
        s_u[r][16 + j] = ou;
    }
    __syncthreads();

    // ---- Phase 2: stream all outputs as 4-wide column groups with b128 NT
    // stores. Lanes cover consecutive groups of one row -> 512B contiguous
    // bursts per wave; NT keeps the 365MB stream from thrashing the 192MB L2.
    // Group boundaries: cols [0,136) are LDS copies, [136,696) gathered admins
    // (16 and 136 are multiples of 4, so each group is homogeneous).
    for (int w = t; w < RPB * NG4; w += TPB) {
        const int r   = w / NG4;
        const int g   = w - r * NG4;
        const int col = g * 4;
        const int row = row0 + r;
        if (row >= batch) continue;

        v4f vl, vu;
        if (col < PRE) {                       // sizes 1 and 2: already in LDS
            vl = *(const v4f*)&s_l[r][col];
            vu = *(const v4f*)&s_u[r][col];
        } else {                               // size 3: 4 gathered admissible-mins
            const int base = col - PRE;
#pragma unroll
            for (int q = 0; q < 4; ++q) {
                uint32_t e = g_tab.lvl3[base + q];
                int il = e & 255, ir = (e >> 8) & 255;
                float ol, ou;
                admin(s_l[r][il], s_u[r][il], s_l[r][ir], s_u[r][ir], ol, ou);
                vl[q] = ol;
                vu[q] = ou;
            }
        }
        const size_t ob = (size_t)row * OUTC + col;   // 16B-aligned (696*4 and col*4 are)
        __builtin_nontemporal_store(vl, (v4f*)&outl[ob]);
        __builtin_nontemporal_store(vu, (v4f*)&outu[ob]);
    }
}

extern "C" void kernel_launch(void* const* d_in, const int* in_sizes, int n_in,
                              void* d_out, int out_size, void* d_ws, size_t ws_size,
                              hipStream_t stream) {
    (void)n_in; (void)out_size; (void)d_ws; (void)ws_size;
    const float* xl = (const float*)d_in[0];
    const float* xu = (const float*)d_in[1];
    const int batch = in_sizes[0] / NCOLS;

    float* outl = (float*)d_out;
    float* outu = outl + (size_t)batch * OUTC;

    const int blocks = (batch + RPB - 1) / RPB;
    min_interval_kernel<<<blocks, TPB, 0, stream>>>(xl, xu, outl, outu, batch);
}